// StackMemory_34608846471638
// MI455X (gfx1250) — compile-verified
//
#include <hip/hip_runtime.h>
#include <math.h>

typedef float v2f __attribute__((ext_vector_type(2)));
typedef float v4f __attribute__((ext_vector_type(4)));
typedef float v8f __attribute__((ext_vector_type(8)));

#define NHEAD 16
#define HDIM 128
#define SDIM 64
#define SLOTS 16
#define DMODEL 2048
#define NEGBIG (-1e9f)
#define LOG_COLS 48           // 3*NHEAD
#define KSPLIT 8

__device__ __forceinline__ v8f wmma_f32(v2f a, v2f b, v8f c) {
  // D = A(16x4 f32) * B(4x16 f32) + C(16x16 f32), exact fp32 path on gfx1250
  return __builtin_amdgcn_wmma_f32_16x16x4_f32(false, a, false, b, (short)0, c, false, false);
}

// ---------------------------------------------------------------------------
// Kernel A: split-K partials of logits = X[N,2048] @ W_action^T[2048,48]
// one wave per (mtile, ksplit); 3 N-tiles per wave (48 cols = 3x16)
// ---------------------------------------------------------------------------
__global__ __launch_bounds__(256) void logits_gemm_partial(
    const float* __restrict__ X, const float* __restrict__ Wa,
    float* __restrict__ part, int ntok) {
  const int lane  = threadIdx.x & 31;
  const int wid   = threadIdx.x >> 5;
  const int gwave = blockIdx.x * 8 + wid;
  const int mtile = gwave >> 3;          // 0..255
  const int ks    = gwave & (KSPLIT - 1);
  const int row16 = lane & 15;
  const int khalf = (lane >> 4) * 2;     // 0 or 2
  const int k0 = ks * (DMODEL / KSPLIT);
  const int k1 = k0 + (DMODEL / KSPLIT);

  const float* arow = X  + (size_t)(mtile * 16 + row16) * DMODEL + khalf;
  const float* b0   = Wa + (size_t)(0 * 16 + row16) * DMODEL + khalf;
  const float* b1   = Wa + (size_t)(1 * 16 + row16) * DMODEL + khalf;
  const float* b2   = Wa + (size_t)(2 * 16 + row16) * DMODEL + khalf;

  v8f c0 = {}; v8f c1 = {}; v8f c2 = {};
  #pragma unroll 4
  for (int kk = k0; kk < k1; kk += 4) {
    v2f a = *(const v2f*)(arow + kk);
    c0 = wmma_f32(a, *(const v2f*)(b0 + kk), c0);
    c1 = wmma_f32(a, *(const v2f*)(b1 + kk), c1);
    c2 = wmma_f32(a, *(const v2f*)(b2 + kk), c2);
  }
  float* p = part + (size_t)ks * ntok * LOG_COLS;
  #pragma unroll
  for (int r = 0; r < 8; ++r) {
    int m   = r + ((lane >> 4) << 3);
    int row = mtile * 16 + m;
    p[row * LOG_COLS +  0 + row16] = c0[r];
    p[row * LOG_COLS + 16 + row16] = c1[r];
    p[row * LOG_COLS + 32 + row16] = c2[r];
  }
}

// deterministic fixed-order split-K reduction + 1/sqrt(HD) scaling
__global__ __launch_bounds__(256) void logits_reduce(
    const float* __restrict__ part, float* __restrict__ logits, int total) {
  int i = blockIdx.x * 256 + threadIdx.x;
  if (i >= total) return;
  float s = 0.f;
  #pragma unroll
  for (int ks = 0; ks < KSPLIT; ++ks) s += part[(size_t)ks * total + i];
  logits[i] = s * 0.08838834764831845f;   // 1/sqrt(128)
}

// ---------------------------------------------------------------------------
// Kernel F: fused per-token kernel. Block = 512 threads = 16 waves = 1 token.
//  phase1: waves 0-3 WMMA  k = X_t(16x128) @ W_down^T  -> LDS
//  phase2: wave h = head: action softmax, stack blend, masked slot softmax,
//          memory_output -> LDS; writes new_stack / new_mask
//  phase3: waves 0-7 WMMA  out = MO(16x64) @ W_up^T, * res_w + hidden
// Stack stream (536 MB round trip) > 192 MB L2 -> non-temporal loads/stores
// so hidden_states (32 MB, read twice) + weights stay L2-resident.
// ---------------------------------------------------------------------------
__global__ __launch_bounds__(512) void stack_fused(
    const float* __restrict__ X, const float* __restrict__ stack_in,
    const float* __restrict__ mask_in, const float* __restrict__ logits,
    const float* __restrict__ Wg, const float* __restrict__ Wd,
    const float* __restrict__ Wu, const float* __restrict__ resw,
    float* __restrict__ out, float* __restrict__ new_stack,
    float* __restrict__ new_mask) {
  __shared__ float k_lds[16 * 68];    // stride 68 -> conflict-free WMMA A reads
  __shared__ float mo_lds[16 * 68];

  const int t     = blockIdx.x;
  const int wave  = threadIdx.x >> 5;
  const int lane  = threadIdx.x & 31;
  const int row16 = lane & 15;
  const int khalf = (lane >> 4) * 2;

  // ---- phase 1: down-projection k (heads = WMMA rows) ----
  if (wave < 4) {
    const float* arow = X  + (size_t)t * DMODEL + row16 * HDIM + khalf;
    const float* brow = Wd + (wave * 16 + row16) * HDIM + khalf;
    v8f c = {};
    #pragma unroll 8
    for (int kk = 0; kk < HDIM; kk += 4)
      c = wmma_f32(*(const v2f*)(arow + kk), *(const v2f*)(brow + kk), c);
    #pragma unroll
    for (int r = 0; r < 8; ++r) {
      int m = r + ((lane >> 4) << 3);
      k_lds[m * 68 + wave * 16 + row16] = c[r];
    }
  }
  __syncthreads();

  // ---- phase 2: stack update (wave = head, lane owns d0=2*lane, d0+1) ----
  {
    const int h  = wave;
    const int d0 = lane * 2;

    const float* lg = logits + (size_t)t * LOG_COLS + 3 * h;
    float l0 = lg[0], l1 = lg[1], l2 = lg[2];
    float mx = fmaxf(l0, fmaxf(l1, l2));
    float e0 = __expf(l0 - mx), e1 = __expf(l1 - mx), e2 = __expf(l2 - mx);
    float inv = 1.0f / (e0 + e1 + e2);
    float a_push = e0 * inv, a_pop = e1 * inv, a_noop = e2 * inv;

    const size_t sbase = ((size_t)t * NHEAD + h) * (SLOTS * SDIM);
    const size_t mbase = ((size_t)t * NHEAD + h) * SLOTS;

    float m[16];
    #pragma unroll
    for (int i = 0; i < 4; ++i) {
      v4f mv = __builtin_nontemporal_load((const v4f*)(mask_in + mbase) + i);
      m[4*i] = mv.x; m[4*i+1] = mv.y; m[4*i+2] = mv.z; m[4*i+3] = mv.w;
    }
    float nm[16];
    #pragma unroll
    for (int s = 0; s < 16; ++s) {
      float pm = (s == 0)  ? 1.0f : m[s - (s ? 1 : 0)];
      float qm = (s == 15) ? 0.0f : m[(s < 15) ? s + 1 : 15];
      nm[s] = a_push * pm + a_pop * qm + a_noop * m[s];
    }
    if (lane < 16)
      __builtin_nontemporal_store(nm[lane], new_mask + mbase + lane);

    v2f st[16];
    #pragma unroll
    for (int s = 0; s < 16; ++s)
      st[s] = __builtin_nontemporal_load((const v2f*)(stack_in + sbase + s * SDIM + d0));
    v2f kv = *(const v2f*)(&k_lds[h * 68 + d0]);
    const v2f zero = {0.0f, 0.0f};

    v2f ns[16];
    #pragma unroll
    for (int s = 0; s < 16; ++s) {
      v2f pu = kv;   if (s > 0)  pu = st[s - 1];
      v2f po = zero; if (s < 15) po = st[s + 1];
      ns[s].x = a_push * pu.x + a_pop * po.x + a_noop * st[s].x;
      ns[s].y = a_push * pu.y + a_pop * po.y + a_noop * st[s].y;
      __builtin_nontemporal_store(ns[s], (v2f*)(new_stack + sbase + s * SDIM + d0));
    }

    v2f wg = *(const v2f*)(Wg + d0);
    float sc[16];
    #pragma unroll
    for (int s = 0; s < 16; ++s) {
      float p = ns[s].x * wg.x + ns[s].y * wg.y;
      #pragma unroll
      for (int off = 16; off >= 1; off >>= 1) p += __shfl_xor(p, off, 32);
      sc[s] = p + (1.0f - nm[s]) * NEGBIG;
    }
    float mxs = sc[0];
    #pragma unroll
    for (int s = 1; s < 16; ++s) mxs = fmaxf(mxs, sc[s]);
    float gw[16]; float sum = 0.f;
    #pragma unroll
    for (int s = 0; s < 16; ++s) { gw[s] = __expf(sc[s] - mxs); sum += gw[s]; }
    float isum = 1.0f / sum;
    v2f mo = zero;
    #pragma unroll
    for (int s = 0; s < 16; ++s) {
      float w = gw[s] * isum;
      mo.x += w * ns[s].x; mo.y += w * ns[s].y;
    }
    mo_lds[h * 68 + d0]     = mo.x;
    mo_lds[h * 68 + d0 + 1] = mo.y;
  }
  __syncthreads();

  // ---- phase 3: up-projection + residual ----
  if (wave < 8) {
    const float* brow = Wu + (wave * 16 + row16) * SDIM + khalf;
    const float* arow = &mo_lds[row16 * 68 + khalf];
    v8f c = {};
    #pragma unroll
    for (int kk = 0; kk < SDIM; kk += 4)
      c = wmma_f32(*(const v2f*)(arow + kk), *(const v2f*)(brow + kk), c);
    const float rw = resw[0];
    #pragma unroll
    for (int r = 0; r < 8; ++r) {
      int m = r + ((lane >> 4) << 3);                 // head
      size_t idx = (size_t)t * DMODEL + m * HDIM + wave * 16 + row16;
      __builtin_nontemporal_store(c[r] * rw + X[idx], out + idx);
    }
  }
}

extern "C" void kernel_launch(void* const* d_in, const int* in_sizes, int n_in,
                              void* d_out, int out_size, void* d_ws, size_t ws_size,
                              hipStream_t stream) {
  const float* X        = (const float*)d_in[0];
  const float* stack_in = (const float*)d_in[1];
  const float* mask_in  = (const float*)d_in[2];
  const float* Wa       = (const float*)d_in[3];
  const float* Wg       = (const float*)d_in[4];
  const float* Wd       = (const float*)d_in[5];
  const float* Wu       = (const float*)d_in[6];
  const float* resw     = (const float*)d_in[7];

  const int ntok = in_sizes[0] / DMODEL;             // B*S = 4096
  float* out       = (float*)d_out;
  float* new_stack = out + (size_t)ntok * DMODEL;
  float* new_mask  = new_stack + (size_t)ntok * NHEAD * SLOTS * SDIM;

  float* logits = (float*)d_ws;                       // ntok*48 floats
  float* part   = logits + (size_t)ntok * LOG_COLS;   // KSPLIT * ntok*48 floats

  // A: 256 mtiles * 8 ksplits = 2048 waves -> 256 blocks of 8 waves
  logits_gemm_partial<<<(ntok / 16) * KSPLIT / 8, 256, 0, stream>>>(X, Wa, part, ntok);
  const int total = ntok * LOG_COLS;
  logits_reduce<<<(total + 255) / 256, 256, 0, stream>>>(part, logits, total);
  stack_fused<<<ntok, 512, 0, stream>>>(X, stack_in, mask_in, logits, Wg, Wd, Wu,
                                        resw, out, new_stack, new_mask);
}